// BKTNN_75582834475151
// MI455X (gfx1250) — compile-verified
//
#include <hip/hip_runtime.h>
#include <hip/hip_bf16.h>
#include <math.h>

// Problem constants (match reference)
#define HD   128
#define BB   128
#define TT   1024
#define EPS_F 1e-6f

#define WAVES    8          // waves per block in mlp kernel
#define HSTRIDE  132        // LDS row stride (floats) -> conflict-free frag loads

typedef __attribute__((ext_vector_type(2))) float v2f;
typedef __attribute__((ext_vector_type(8))) float v8f;

struct MlpP { const float* W[5]; const float* b[5]; };

__device__ __forceinline__ float sigmoidf_(float x) { return 1.0f / (1.0f + expf(-x)); }
__device__ __forceinline__ float clipf_(float x)    { return fminf(fmaxf(x, EPS_F), 1.0f - EPS_F); }

// ---------------------------------------------------------------------------
// Cooperative load of a 128x128 row-major weight matrix into LDS, pre-swizzled
// into B-fragment order: Wsw[(k*8+ct)*64 + lane*2 + v] = W[4k + rv][16ct + cn]
// where lane = cn + 16*(rv>=2), v = rv&1.  B-frag read is then a contiguous,
// bank-conflict-free ds_load_b64 per lane.
// ---------------------------------------------------------------------------
__device__ __forceinline__ void load_w_swz(float* Wsw, float* bsh,
                                           const float* __restrict__ Wg,
                                           const float* __restrict__ bg,
                                           int tid) {
    for (int i = tid; i < HD * HD; i += 256) {
        int row = i >> 7, col = i & 127;
        int k  = row >> 2, rv = row & 3;
        int ct = col >> 4, cn = col & 15;
        int lane = cn + ((rv >> 1) << 4);
        int v    = rv & 1;
        Wsw[(k * 8 + ct) * 64 + lane * 2 + v] = Wg[i];
    }
    if (tid < HD) bsh[tid] = bg[tid];
}

// Head layer: 128x3 -> padded to 128x16 (cols 3..15 zero), same swizzle, ct=0.
__device__ __forceinline__ void load_w5_swz(float* Wsw, float* bsh,
                                            const float* __restrict__ Wg,
                                            const float* __restrict__ bg,
                                            int tid) {
    for (int i = tid; i < 32 * 64; i += 256) {
        int k = i >> 6, r = i & 63;
        int lane = r >> 1, v = r & 1;
        int cn = lane & 15;
        int rv = ((lane >> 4) << 1) + v;
        int row = 4 * k + rv;
        Wsw[i] = (cn < 3) ? Wg[row * 3 + cn] : 0.0f;
    }
    if (tid < 16) bsh[tid] = (tid < 3) ? bg[tid] : 0.0f;
}

// ---------------------------------------------------------------------------
// One dense layer for a 16-row tile:  h(16x128) = act(h @ W + b)  in place.
// A-frag: lane m<16 holds row m cols {4k,4k+1}; lane m>=16 holds row m-16
// cols {4k+2,4k+3}.  D: lane L holds col (L&15), rows v + 8*(L>=16).
// ---------------------------------------------------------------------------
__device__ __forceinline__ void layer16(float* __restrict__ h,
                                        const float* __restrict__ Wsw,
                                        const float* __restrict__ bsh,
                                        int lane, bool do_relu) {
    const int c     = lane & 15;
    const int hiK   = (lane >> 4) << 1;   // 0 or 2
    const int mbase = (lane >> 4) << 3;   // 0 or 8

    v8f acc[8];
#pragma unroll
    for (int ct = 0; ct < 8; ++ct) {
        float bv = bsh[ct * 16 + c];
        acc[ct] = (v8f){bv, bv, bv, bv, bv, bv, bv, bv};
    }

    for (int k = 0; k < 32; ++k) {
        v2f a = *(const v2f*)(h + c * HSTRIDE + 4 * k + hiK);
#pragma unroll
        for (int ct = 0; ct < 8; ++ct) {
            v2f bf = *(const v2f*)(Wsw + (k * 8 + ct) * 64 + lane * 2);
            acc[ct] = __builtin_amdgcn_wmma_f32_16x16x4_f32(
                false, a, false, bf, (short)0, acc[ct], false, false);
        }
    }

#pragma unroll
    for (int ct = 0; ct < 8; ++ct) {
#pragma unroll
        for (int v = 0; v < 8; ++v) {
            float x = acc[ct][v];
            if (do_relu) x = fmaxf(x, 0.0f);
            h[(mbase + v) * HSTRIDE + ct * 16 + c] = x;
        }
    }
}

// Head layer (single 16-col tile), returns pre-sigmoid 16x16 D fragment.
__device__ __forceinline__ v8f head16(const float* __restrict__ h,
                                      const float* __restrict__ Wsw,
                                      const float* __restrict__ bsh,
                                      int lane) {
    const int c   = lane & 15;
    const int hiK = (lane >> 4) << 1;
    float bv = bsh[c];
    v8f acc = (v8f){bv, bv, bv, bv, bv, bv, bv, bv};
    for (int k = 0; k < 32; ++k) {
        v2f a  = *(const v2f*)(h + c * HSTRIDE + 4 * k + hiK);
        v2f bf = *(const v2f*)(Wsw + k * 64 + lane * 2);
        acc = __builtin_amdgcn_wmma_f32_16x16x4_f32(
            false, a, false, bf, (short)0, acc, false, false);
    }
    return acc;
}

// ---------------------------------------------------------------------------
// Main fused kernel: gather embeddings, run both MLPs, emit clipped (l,g,s).
// Row order r = t*BB + b  (scan-friendly).  params layout: [T][B][3].
// ---------------------------------------------------------------------------
__global__ __launch_bounds__(256) void bkt_mlp_kernel(
    const int* __restrict__ X,
    MlpP sk, MlpP mx,
    const float* __restrict__ semb,
    const float* __restrict__ memb,
    float* __restrict__ params) {
    __shared__ __align__(16) float Wsw[32 * 8 * 64];        // 64 KB
    __shared__ __align__(16) float bsh[HD];
    __shared__ __align__(16) float hsh[WAVES * 16 * HSTRIDE];

    const int tid  = threadIdx.x;
    const int lane = tid & 31;
    const int wave = tid >> 5;
    float* h = hsh + wave * (16 * HSTRIDE);
    const int rr0 = blockIdx.x * (WAVES * 16) + wave * 16;

    const int c     = lane & 15;
    const int mbase = (lane >> 4) << 3;

    // ---- skill branch: h = skill_emb[X[...,0]] ----
    for (int j = 0; j < 16; ++j) {
        int r = rr0 + j;
        int b = r & (BB - 1);
        int t = r >> 7;
        int is = X[(b * TT + t) * 2];
        if (is < 0) is = 0;
        float4 v = *(const float4*)(semb + is * HD + lane * 4);
        *(float4*)(h + j * HSTRIDE + lane * 4) = v;
    }
    for (int l = 0; l < 4; ++l) {
        __syncthreads();
        load_w_swz(Wsw, bsh, sk.W[l], sk.b[l], tid);
        __syncthreads();
        layer16(h, Wsw, bsh, lane, true);
    }
    __syncthreads();
    load_w5_swz(Wsw, bsh, sk.W[4], sk.b[4], tid);
    __syncthreads();
    {
        v8f d = head16(h, Wsw, bsh, lane);
        if (c == 0) {  // column 0 -> l
#pragma unroll
            for (int v = 0; v < 8; ++v)
                params[(rr0 + mbase + v) * 3 + 0] = clipf_(sigmoidf_(d[v]));
        }
    }

    // ---- mplex branch: h = skill_emb[X0] + mplex_emb[X1] ----
    for (int j = 0; j < 16; ++j) {
        int r = rr0 + j;
        int b = r & (BB - 1);
        int t = r >> 7;
        int is = X[(b * TT + t) * 2];     if (is < 0) is = 0;
        int im = X[(b * TT + t) * 2 + 1]; if (im < 0) im = 0;
        float4 a = *(const float4*)(semb + is * HD + lane * 4);
        float4 m = *(const float4*)(memb + im * HD + lane * 4);
        a.x += m.x; a.y += m.y; a.z += m.z; a.w += m.w;
        *(float4*)(h + j * HSTRIDE + lane * 4) = a;
    }
    for (int l = 0; l < 4; ++l) {
        __syncthreads();
        load_w_swz(Wsw, bsh, mx.W[l], mx.b[l], tid);
        __syncthreads();
        layer16(h, Wsw, bsh, lane, true);
    }
    __syncthreads();
    load_w5_swz(Wsw, bsh, mx.W[4], mx.b[4], tid);
    __syncthreads();
    {
        v8f d = head16(h, Wsw, bsh, lane);
        if (c == 1 || c == 2) {  // columns 1,2 -> g,s
#pragma unroll
            for (int v = 0; v < 8; ++v)
                params[(rr0 + mbase + v) * 3 + c] = clipf_(sigmoidf_(d[v]));
        }
    }
}

// ---------------------------------------------------------------------------
// Prior MLP on se[:,0] : tiny (128 rows), one block per batch element.
// ---------------------------------------------------------------------------
__global__ __launch_bounds__(128) void bkt_prior_kernel(
    const int* __restrict__ X, const float* __restrict__ semb,
    const float* __restrict__ W1, const float* __restrict__ b1,
    const float* __restrict__ W2, const float* __restrict__ b2,
    float* __restrict__ latent0) {
    __shared__ float se[HD];
    __shared__ float red[HD];
    const int b   = blockIdx.x;
    const int tid = threadIdx.x;
    int idx = X[(b * TT + 0) * 2];
    if (idx < 0) idx = 0;
    se[tid] = semb[idx * HD + tid];
    __syncthreads();
    float acc = b1[tid];
    for (int k = 0; k < HD; ++k) acc = fmaf(se[k], W1[k * HD + tid], acc);
    float h1 = fmaxf(acc, 0.0f);
    red[tid] = h1 * W2[tid];
    __syncthreads();
    for (int s = 64; s > 0; s >>= 1) {
        if (tid < s) red[tid] += red[tid + s];
        __syncthreads();
    }
    if (tid == 0) latent0[b] = sigmoidf_(red[0] + b2[0]);
}

// ---------------------------------------------------------------------------
// Sequential BKT scan: one thread per batch lane, coalesced param reads.
// out[0:B*T) = corrects (B,T row-major), out[B*T:2BT) = latents.
// ---------------------------------------------------------------------------
__global__ __launch_bounds__(128) void bkt_scan_kernel(
    const float* __restrict__ params, const float* __restrict__ latent0,
    const int* __restrict__ y, float* __restrict__ out) {
    const int b = threadIdx.x;
    float latent = latent0[b];
    float* corrects = out;
    float* latents  = out + BB * TT;
    for (int t = 0; t < TT; ++t) {
        const float* p = params + (t * BB + b) * 3;
        float l = p[0], g = p[1], s = p[2];
        int yt = y[b * TT + t];           // y==-1 behaves as 0
        float om = 1.0f - latent;
        float correct = latent * (1.0f - s) + om * g;
        float kt;
        if (yt > 0) kt = latent * (1.0f - s) / correct;
        else        kt = latent * s / (latent * s + om * (1.0f - g));
        corrects[b * TT + t] = correct;
        latents[b * TT + t]  = latent;
        latent = fminf(fmaxf(kt + (1.0f - kt) * l, EPS_F), 1.0f - EPS_F);
    }
}

// ---------------------------------------------------------------------------
extern "C" void kernel_launch(void* const* d_in, const int* in_sizes, int n_in,
                              void* d_out, int out_size, void* d_ws, size_t ws_size,
                              hipStream_t stream) {
    // setup_inputs order: X, y, skill_mlp{Ws[5],bs[5]}, mplex_mlp{Ws[5],bs[5]},
    // prior_mlp{Ws[2],bs[2]}, skill_emb, mplex_emb
    const int* X = (const int*)d_in[0];
    const int* y = (const int*)d_in[1];
    MlpP sk, mx;
    for (int i = 0; i < 5; ++i) {
        sk.W[i] = (const float*)d_in[2 + i];
        sk.b[i] = (const float*)d_in[7 + i];
        mx.W[i] = (const float*)d_in[12 + i];
        mx.b[i] = (const float*)d_in[17 + i];
    }
    const float* pW1 = (const float*)d_in[22];
    const float* pW2 = (const float*)d_in[23];
    const float* pb1 = (const float*)d_in[24];
    const float* pb2 = (const float*)d_in[25];
    const float* semb = (const float*)d_in[26];
    const float* memb = (const float*)d_in[27];

    float* params  = (float*)d_ws;               // [T][B][3]
    float* latent0 = params + (size_t)BB * TT * 3;
    float* out     = (float*)d_out;

    const int rows = BB * TT;
    bkt_mlp_kernel<<<rows / (WAVES * 16), 256, 0, stream>>>(X, sk, mx, semb, memb, params);
    bkt_prior_kernel<<<BB, 128, 0, stream>>>(X, semb, pW1, pb1, pW2, pb2, latent0);
    bkt_scan_kernel<<<1, BB, 0, stream>>>(params, latent0, y, out);
}